// SpatialDIVA_35682588295501
// MI455X (gfx1250) — compile-verified
//
#include <hip/hip_runtime.h>
#include <hip/hip_bf16.h>
#include <math.h>

#define BB   4096
#define DD   4000
#define KK   6
#define ZZ   64
#define FEPS 1e-8f

typedef float  f4  __attribute__((ext_vector_type(4)));
typedef int    i4  __attribute__((ext_vector_type(4)));
typedef float  v2f __attribute__((ext_vector_type(2)));
typedef float  v8f __attribute__((ext_vector_type(8)));

// ---------------------------------------------------------------------------
// Kernel 1: fused per-row pass. One block per row b. Streams all [B,D] arrays
// and the [B,K,D] neighbor tensor once with non-temporal b128 loads, computes
//   accN = sum_d ( mask[d] ? 0.5*(x-mu_g)^2 : -ll_nb )
//   accS = sum_d ( qy_loc - (x_input - mean_k neighbor) )^2
// and writes per-block partials to ws[b] / ws[BB+b].
// ---------------------------------------------------------------------------
__global__ __launch_bounds__(256) void fused_nll_spatial_kernel(
    const float* __restrict__ x,
    const float* __restrict__ mu_g,
    const float* __restrict__ mu_nb,
    const float* __restrict__ th_nb,
    const float* __restrict__ x_in,
    const float* __restrict__ nbr,
    const float* __restrict__ qy,
    const int*   __restrict__ mask,
    float*       __restrict__ ws)
{
    const int b = blockIdx.x;
    const int t = threadIdx.x;
    const size_t ro = (size_t)b * DD;        // row offset, 16B-aligned (DD%4==0)
    const size_t no = (size_t)b * KK * DD;   // neighbor block offset

    float accN = 0.0f;
    float accS = 0.0f;

    const int nvec = DD / 4;                 // 1000
    for (int v = t; v < nvec; v += 256) {
        const int d = v * 4;
        f4 xv  = __builtin_nontemporal_load((const f4*)(x     + ro + d));
        f4 mgv = __builtin_nontemporal_load((const f4*)(mu_g  + ro + d));
        f4 mnv = __builtin_nontemporal_load((const f4*)(mu_nb + ro + d));
        f4 thv = __builtin_nontemporal_load((const f4*)(th_nb + ro + d));
        f4 xiv = __builtin_nontemporal_load((const f4*)(x_in  + ro + d));
        f4 qv  = __builtin_nontemporal_load((const f4*)(qy    + ro + d));
        i4 mv  = *(const i4*)(mask + d);     // tiny, cache-resident

        f4 ns = {0.0f, 0.0f, 0.0f, 0.0f};
        #pragma unroll
        for (int k = 0; k < KK; ++k) {
            f4 nv = __builtin_nontemporal_load((const f4*)(nbr + no + (size_t)k * DD + d));
            ns += nv;
        }

        #pragma unroll
        for (int j = 0; j < 4; ++j) {
            // spatial term: neighbor_diffs = x_input - mean_k(neighbor)
            float nd = xiv[j] - ns[j] * (1.0f / (float)KK);
            float sd = qv[j] - nd;
            accS = fmaf(sd, sd, accS);

            if (mv[j]) {
                float dd = xv[j] - mgv[j];
                accN = fmaf(0.5f * dd, dd, accN);
            } else {
                float th = thv[j], mu = mnv[j], xx = xv[j];
                float ltme = logf(th + mu + FEPS);
                float ll = th * (logf(th + FEPS) - ltme)
                         + xx * (logf(mu + FEPS) - ltme)
                         + lgammaf(xx + th) - lgammaf(th) - lgammaf(xx + 1.0f);
                accN -= ll;
            }
        }
    }

    __shared__ float sN[256];
    __shared__ float sS[256];
    sN[t] = accN;
    sS[t] = accS;
    __syncthreads();
    #pragma unroll
    for (int s = 128; s > 0; s >>= 1) {
        if (t < s) { sN[t] += sN[t + s]; sS[t] += sS[t + s]; }
        __syncthreads();
    }
    if (t == 0) {
        ws[b]      = sN[0];
        ws[BB + b] = sS[0];
    }
}

// ---------------------------------------------------------------------------
// Kernel 2: single-wave deterministic finalize using V_WMMA_F32_16X16X4_F32
// as the cross-lane reducer. A-matrix 16x4 layout: lane m (0-15) VGPR0 =
// A[m][0], lanes 16-31 VGPR0 = A[m][2]; VGPR1 slots set to 0. B = all-ones,
// so D[m][n] = v_m + v_{m+16} (= r_m) for every n. Row sums land in D VGPRs
// 0..7 (r_0..r_7 in lanes 0-15, r_8..r_15 in lanes 16-31); 8-way add +
// shfl_xor(16) finishes. Launched with exactly 32 threads: one wave32, EXEC
// all ones, uniform control flow up to the WMMAs.
// ---------------------------------------------------------------------------
__device__ __forceinline__ float wave_reduce_wmma(float v, int lane) {
    v2f A;  A.x = v;    A.y = 0.0f;
    v2f Bm; Bm.x = 1.0f; Bm.y = 1.0f;
    v8f C = {0.0f, 0.0f, 0.0f, 0.0f, 0.0f, 0.0f, 0.0f, 0.0f};
    v8f Dm = __builtin_amdgcn_wmma_f32_16x16x4_f32(
        /*neg_a=*/false, A, /*neg_b=*/false, Bm,
        /*c_mod=*/(short)0, C, /*reuse_a=*/false, /*reuse_b=*/false);
    float s = Dm[0] + Dm[1] + Dm[2] + Dm[3] + Dm[4] + Dm[5] + Dm[6] + Dm[7];
    s += __shfl_xor(s, 16, 32);   // combine halves: total in every lane
    (void)lane;
    return s;
}

__global__ __launch_bounds__(32) void finalize_kernel(
    const float* __restrict__ ws, float* __restrict__ out)
{
    const int lane = threadIdx.x;
    float aN = 0.0f, aS = 0.0f;
    #pragma unroll 4
    for (int i = lane; i < BB; i += 32) {   // uniform trip count: EXEC stays full
        aN += ws[i];
        aS += ws[BB + i];
    }
    float totN = wave_reduce_wmma(aN, lane);
    float totS = wave_reduce_wmma(aS, lane);
    if (lane == 0) {
        out[0] = totN / (float)BB;                     // nll_total (mean over B)
        out[1] = totS / ((float)BB * (float)DD);       // spatial_loss (mean over B*D)
    }
}

// ---------------------------------------------------------------------------
// Kernel 3: reparametrize z = mu + eps * exp(0.5*logvar), [B,Z] = 262144 elems
// ---------------------------------------------------------------------------
__global__ __launch_bounds__(256) void reparam_kernel(
    const float* __restrict__ mu_z,
    const float* __restrict__ lv_z,
    const float* __restrict__ eps,
    float*       __restrict__ z)
{
    const int i = blockIdx.x * 256 + threadIdx.x;
    if (i < BB * ZZ) {
        z[i] = fmaf(eps[i], __expf(0.5f * lv_z[i]), mu_z[i]);
    }
}

extern "C" void kernel_launch(void* const* d_in, const int* in_sizes, int n_in,
                              void* d_out, int out_size, void* d_ws, size_t ws_size,
                              hipStream_t stream)
{
    (void)in_sizes; (void)n_in; (void)out_size; (void)ws_size;

    const float* x       = (const float*)d_in[0];
    const float* mu_g    = (const float*)d_in[1];
    // d_in[2] = logvar_gauss: unused (reference uses unit variance)
    const float* mu_nb   = (const float*)d_in[3];
    const float* th_nb   = (const float*)d_in[4];
    const float* x_in    = (const float*)d_in[5];
    const float* nbr     = (const float*)d_in[6];
    const float* qy      = (const float*)d_in[7];
    const float* mu_z    = (const float*)d_in[8];
    const float* lv_z    = (const float*)d_in[9];
    const float* eps     = (const float*)d_in[10];
    const int*   mask    = (const int*)d_in[11];

    float* out = (float*)d_out;       // [0]=nll, [1]=spatial, [2..] = z (B*Z)
    float* ws  = (float*)d_ws;        // 2*BB floats of per-row partials

    fused_nll_spatial_kernel<<<BB, 256, 0, stream>>>(
        x, mu_g, mu_nb, th_nb, x_in, nbr, qy, mask, ws);

    finalize_kernel<<<1, 32, 0, stream>>>(ws, out);

    reparam_kernel<<<(BB * ZZ) / 256, 256, 0, stream>>>(
        mu_z, lv_z, eps, out + 2);
}